// FourierQTCN_2104533975408
// MI455X (gfx1250) — compile-verified
//
#include <hip/hip_runtime.h>
#include <math.h>

#define NQ 10
#define WDIM 113
#define BATCH 16
#define CCH 22
#define TLEN 128
#define KS 16
#define NF 8
#define FDIM (CCH*NF*2)   /* 352 */
#define BM (BATCH*WDIM)   /* 1808 = 113*16 */
#define NGATES 23

typedef __attribute__((ext_vector_type(2))) float v2f;
typedef __attribute__((ext_vector_type(8))) float v8f;

struct c32 { float x, y; };
__device__ inline c32 cmul(c32 a, c32 b) { return c32{a.x*b.x - a.y*b.y, a.x*b.y + a.y*b.x}; }
__device__ inline c32 cmac(c32 acc, c32 a, c32 b) {
    return c32{ fmaf(a.x, b.x, fmaf(-a.y, b.y, acc.x)),
                fmaf(a.x, b.y, fmaf( a.y, b.x, acc.y)) };
}

// ---- circuit topology (program order of the 23 fused 2q gates) ----
// type: 0 = fused conv pair, 1 = pool cu3
__constant__ int g_type[NGATES]  = {0,0,0,0,0, 0,0,0,0, 1,1,1,1,1, 0,0,0,0, 1,1, 0,0, 1};
__constant__ int g_layer[NGATES] = {0,0,0,0,0, 0,0,0,0, 0,0,0,0,0, 1,1,1,1, 1,1, 2,2, 2};
__constant__ int g_idx[NGATES]   = {0,2,4,6,8, 1,3,5,7, 0,1,2,3,4, 0,2,1,3, 0,1, 0,1, 0};
__constant__ int g_w1[NGATES]    = {0,2,4,6,8, 1,3,5,7, 1,3,5,7,9, 0,4,2,6, 2,6, 0,4, 4};
__constant__ int g_w2[NGATES]    = {1,3,5,7,9, 2,4,6,8, 0,2,4,6,8, 2,6,4,8, 0,4, 4,8, 0};

// =============== Kernel A: windowed 16-pt DFT features ===============
__global__ void __launch_bounds__(256) feat_kernel(const float* __restrict__ x,
                                                   float* __restrict__ feats) {
    __shared__ c32 tw[NF][KS];
    int tid = threadIdx.x;
    if (tid < NF*KS) {
        int k = tid / KS, n = tid % KS;
        float ang = -2.0f * 3.14159265358979f * (float)(k*n) / 16.0f;
        tw[k][n] = c32{cosf(ang), sinf(ang)};
    }
    __syncthreads();
    int g = blockIdx.x * 256 + tid;
    if (g >= BATCH*CCH*WDIM) return;
    int w = g % WDIM;
    int c = (g / WDIM) % CCH;
    int b = g / (WDIM*CCH);
    const float* xp = x + (b*CCH + c)*TLEN + w;
    float xv[KS];
    #pragma unroll
    for (int n = 0; n < KS; ++n) xv[n] = xp[n];
    float* fr = feats + (size_t)(b*WDIM + w)*FDIM + c*(NF*2);
    #pragma unroll
    for (int k = 0; k < NF; ++k) {
        float re = 0.f, im = 0.f;
        #pragma unroll
        for (int n = 0; n < KS; ++n) {
            re = fmaf(xv[n], tw[k][n].x, re);
            im = fmaf(xv[n], tw[k][n].y, im);
        }
        float mag = sqrtf(re*re + im*im);
        fr[2*k+0] = log1pf(mag);
        fr[2*k+1] = atan2f(im, re) * 0.318309886183790672f; // 1/pi
    }
}

// =============== Kernel A2: pad proj_w (10x352) -> pwpad (16x352, zero rows 10..15) ===============
__global__ void __launch_bounds__(256) pad_kernel(const float* __restrict__ pw,
                                                  float* __restrict__ pwpad) {
    int i = blockIdx.x * 256 + threadIdx.x;
    if (i >= 16*FDIM) return;
    int n = i / FDIM;
    pwpad[i] = (n < NQ) ? pw[i] : 0.f;
}

// =============== Kernel B: projection GEMM via V_WMMA_F32_16X16X4_F32 ===============
// One wave per 16-row M tile. A: 16x4 fp32 (lanes 0-15 = rows, lane[4] selects K pair).
// B: 4x16 (= padded proj_w^T slice), mirrored layout.
// C/D: VGPR r -> M=r (lanes 0-15) / M=r+8 (lanes 16-31), N = lane&15.
__global__ void __launch_bounds__(32) proj_wmma_kernel(const float* __restrict__ feats,
                                                       const float* __restrict__ pwpad,
                                                       const float* __restrict__ pb,
                                                       float* __restrict__ angles) {
    int tile = blockIdx.x;        // 0..112
    int lane = threadIdx.x;       // 0..31
    int row   = tile*16 + (lane & 15);
    int khalf = (lane >> 4) * 2;
    int n     = lane & 15;
    const float* ap = feats + (size_t)row*FDIM + khalf;
    const float* bp = pwpad + (size_t)n*FDIM + khalf;
    v8f acc = {};
    for (int k0 = 0; k0 < FDIM; k0 += 4) {
        v2f a, bf;
        a.x  = ap[k0];  a.y  = ap[k0 + 1];
        bf.x = bp[k0];  bf.y = bp[k0 + 1];
        acc = __builtin_amdgcn_wmma_f32_16x16x4_f32(false, a, false, bf,
                                                    (short)0, acc, false, false);
    }
    int col = lane & 15;
    int rbase = (lane >> 4) * 8;
    if (col < NQ) {
        float bias = pb[col];
        #pragma unroll
        for (int r = 0; r < 8; ++r) {
            int m = tile*16 + rbase + r;
            angles[(size_t)m*NQ + col] = tanhf(acc[r] + bias) * 3.14159265358979f;
        }
    }
}

// =============== Kernel C: precompute the 23 fused 4x4 complex gates ===============
__device__ inline void u3mat(float t, float p, float l, c32 u[2][2]) {
    float ct, st, cp, sp, cl, sl;
    sincosf(0.5f*t, &st, &ct);
    sincosf(p, &sp, &cp);
    sincosf(l, &sl, &cl);
    float cpl = cp*cl - sp*sl, spl = sp*cl + cp*sl;   // e^{i(p+l)}
    u[0][0] = c32{ct, 0.f};
    u[0][1] = c32{-cl*st, -sl*st};
    u[1][0] = c32{ cp*st,  sp*st};
    u[1][1] = c32{cpl*ct, spl*ct};
}

__device__ inline void mat4mul(const c32* A, const c32* B, c32* C) {
    for (int i = 0; i < 4; ++i)
        for (int j = 0; j < 4; ++j) {
            c32 s{0.f, 0.f};
            for (int k = 0; k < 4; ++k) s = cmac(s, A[i*4+k], B[k*4+j]);
            C[i*4+j] = s;
        }
}

__global__ void gate_setup_kernel(const float* __restrict__ cp,
                                  const float* __restrict__ pp,
                                  c32* __restrict__ gates) {
    int g = threadIdx.x;
    if (g >= NGATES) return;
    c32 M[16];
    int L = g_layer[g], r = g_idx[g];
    if (g_type[g] == 0) {
        const float* w0 = cp + L*150 + r*15;
        const float* w1 = cp + L*150 + (r+1)*15;
        c32 preW[2][2], preN[2][2], postW[2][2], postN[2][2];
        u3mat(w0[0],  w0[1],  w0[2],  preW);
        u3mat(w1[3],  w1[4],  w1[5],  preN);
        u3mat(w0[9],  w0[10], w0[11], postW);
        u3mat(w1[12], w1[13], w1[14], postN);
        float cz, sz, cy, sy, cx, sx;
        sincosf(0.5f*w0[6], &sz, &cz);
        sincosf(0.5f*w0[7], &sy, &cy);
        sincosf(0.5f*w0[8], &sx, &cx);
        c32 ZZ[16] = {}; c32 YY[16] = {}; c32 XX[16] = {};
        ZZ[0]=c32{cz,-sz}; ZZ[5]=c32{cz,sz}; ZZ[10]=c32{cz,sz}; ZZ[15]=c32{cz,-sz};
        YY[0]=c32{cy,0}; YY[5]=c32{cy,0}; YY[10]=c32{cy,0}; YY[15]=c32{cy,0};
        YY[3]=c32{0, sy}; YY[12]=c32{0, sy}; YY[6]=c32{0,-sy}; YY[9]=c32{0,-sy};
        XX[0]=c32{cx,0}; XX[5]=c32{cx,0}; XX[10]=c32{cx,0}; XX[15]=c32{cx,0};
        XX[3]=c32{0,-sx}; XX[6]=c32{0,-sx}; XX[9]=c32{0,-sx}; XX[12]=c32{0,-sx};
        c32 PRE[16], POST[16];
        for (int a = 0; a < 2; ++a)
            for (int b = 0; b < 2; ++b)
                for (int ci = 0; ci < 2; ++ci)
                    for (int d = 0; d < 2; ++d) {
                        PRE [(2*a+b)*4 + (2*ci+d)] = cmul(preW [a][ci], preN [b][d]);
                        POST[(2*a+b)*4 + (2*ci+d)] = cmul(postW[a][ci], postN[b][d]);
                    }
        c32 T1[16], T2[16], T3[16];
        mat4mul(ZZ, PRE, T1);
        mat4mul(YY, T1, T2);
        mat4mul(XX, T2, T3);
        mat4mul(POST, T3, M);
    } else {
        const float* q = pp + L*15 + r*3;
        c32 u[2][2]; u3mat(q[0], q[1], q[2], u);
        for (int i = 0; i < 16; ++i) M[i] = c32{0.f, 0.f};
        M[0] = c32{1.f, 0.f}; M[5] = c32{1.f, 0.f};
        M[10] = u[0][0]; M[11] = u[0][1];
        M[14] = u[1][0]; M[15] = u[1][1];
    }
    for (int i = 0; i < 16; ++i) gates[g*16 + i] = M[i];
}

// =============== Kernel D: statevector simulation (one WG = one window) ===============
__global__ void __launch_bounds__(256) circuit_kernel(const float* __restrict__ angles,
                                                      const float* __restrict__ fscale,
                                                      const c32* __restrict__ gatesG,
                                                      float* __restrict__ ev) {
    __shared__ c32 st[1024];            // 8 KB statevector
    __shared__ c32 gates[NGATES*16];    // ~3 KB gate cache
    __shared__ c32 vq[NQ][2];
    __shared__ float red[256];
    int t = threadIdx.x;
    int bm = blockIdx.x;

    for (int i = t; i < NGATES*16; i += 256) gates[i] = gatesG[i];
    if (t < NQ) {
        float a = angles[(size_t)bm*NQ + t];
        float c, s, cb, sb;
        sincosf(0.5f*a, &s, &c);
        sincosf(0.5f*fscale[t]*a, &sb, &cb);
        vq[t][0] = c32{cb*c, -sb*s};    // (RX(fs*a) RY(a)) |0>
        vq[t][1] = c32{cb*s, -sb*c};
    }
    __syncthreads();

    // product-state init
    for (int idx = t; idx < 1024; idx += 256) {
        c32 amp{1.f, 0.f};
        #pragma unroll
        for (int q = 0; q < NQ; ++q) amp = cmul(amp, vq[q][(idx >> (9 - q)) & 1]);
        st[idx] = amp;
    }
    __syncthreads();

    #pragma unroll 1
    for (int g = 0; g < NGATES; ++g) {
        int p1 = 9 - g_w1[g], p2 = 9 - g_w2[g];
        int pl = min(p1, p2), ph = max(p1, p2);
        unsigned low = (unsigned)t & ((1u << pl) - 1u);
        unsigned mid = ((unsigned)t >> pl) & ((1u << (ph - 1 - pl)) - 1u);
        unsigned hi  = (unsigned)t >> (ph - 1);
        unsigned base = low | (mid << (pl + 1)) | (hi << (ph + 1));
        unsigned i0 = base;
        unsigned i1 = base | (1u << p2);
        unsigned i2 = base | (1u << p1);
        unsigned i3 = i1 | (1u << p1);
        c32 s0 = st[i0], s1 = st[i1], s2 = st[i2], s3 = st[i3];
        const c32* G = &gates[g*16];
        c32 n0{0,0}, n1{0,0}, n2{0,0}, n3{0,0};
        n0 = cmac(cmac(cmac(cmac(n0, G[ 0], s0), G[ 1], s1), G[ 2], s2), G[ 3], s3);
        n1 = cmac(cmac(cmac(cmac(n1, G[ 4], s0), G[ 5], s1), G[ 6], s2), G[ 7], s3);
        n2 = cmac(cmac(cmac(cmac(n2, G[ 8], s0), G[ 9], s1), G[10], s2), G[11], s3);
        n3 = cmac(cmac(cmac(cmac(n3, G[12], s0), G[13], s1), G[14], s2), G[15], s3);
        st[i0] = n0; st[i1] = n1; st[i2] = n2; st[i3] = n3;
        __syncthreads();
    }

    // ev = P(q0=0) - P(q0=1)  (q0 is the MSB of the flat index)
    float acc = 0.f;
    for (int idx = t; idx < 1024; idx += 256) {
        float m = st[idx].x*st[idx].x + st[idx].y*st[idx].y;
        acc += (idx < 512) ? m : -m;
    }
    red[t] = acc; __syncthreads();
    for (int sft = 128; sft > 0; sft >>= 1) {
        if (t < sft) red[t] += red[t + sft];
        __syncthreads();
    }
    if (t == 0) ev[bm] = red[0];
}

// =============== Kernel E: softmax-weighted aggregation over W ===============
__global__ void __launch_bounds__(128) agg_kernel(const float* __restrict__ aw,
                                                  const float* __restrict__ ev,
                                                  float* __restrict__ out) {
    __shared__ float sm[128];
    int t = threadIdx.x, b = blockIdx.x;
    float v = (t < WDIM) ? aw[t] : -1e30f;
    sm[t] = v; __syncthreads();
    for (int s = 64; s > 0; s >>= 1) { if (t < s) sm[t] = fmaxf(sm[t], sm[t+s]); __syncthreads(); }
    float mx = sm[0]; __syncthreads();
    float e = (t < WDIM) ? __expf(v - mx) : 0.f;
    sm[t] = e; __syncthreads();
    for (int s = 64; s > 0; s >>= 1) { if (t < s) sm[t] += sm[t+s]; __syncthreads(); }
    float denom = sm[0]; __syncthreads();
    float ws = (t < WDIM) ? ev[(size_t)b*WDIM + t] * e : 0.f;
    sm[t] = ws; __syncthreads();
    for (int s = 64; s > 0; s >>= 1) { if (t < s) sm[t] += sm[t+s]; __syncthreads(); }
    if (t == 0) out[b] = sm[0] / denom;
}

extern "C" void kernel_launch(void* const* d_in, const int* in_sizes, int n_in,
                              void* d_out, int out_size, void* d_ws, size_t ws_size,
                              hipStream_t stream) {
    (void)in_sizes; (void)n_in; (void)out_size; (void)ws_size;
    const float* x       = (const float*)d_in[0];
    const float* proj_w  = (const float*)d_in[1];
    const float* proj_b  = (const float*)d_in[2];
    const float* fscale  = (const float*)d_in[3];
    const float* conv_p  = (const float*)d_in[4];
    const float* pool_p  = (const float*)d_in[5];
    const float* agg_w   = (const float*)d_in[6];
    float* out = (float*)d_out;

    char* ws = (char*)d_ws;
    float* feats  = (float*)ws;                                  // BM*FDIM
    float* angles = feats + (size_t)BM*FDIM;                     // BM*NQ
    float* pwpad  = angles + (size_t)BM*NQ;                      // 16*FDIM
    c32*   gates  = (c32*)(pwpad + 16*FDIM);                     // NGATES*16
    float* ev     = (float*)(gates + NGATES*16);                 // BM

    feat_kernel<<<(BATCH*CCH*WDIM + 255)/256, 256, 0, stream>>>(x, feats);
    pad_kernel<<<(16*FDIM + 255)/256, 256, 0, stream>>>(proj_w, pwpad);
    proj_wmma_kernel<<<WDIM, 32, 0, stream>>>(feats, pwpad, proj_b, angles);
    gate_setup_kernel<<<1, 32, 0, stream>>>(conv_p, pool_p, gates);
    circuit_kernel<<<BM, 256, 0, stream>>>(angles, fscale, gates, ev);
    agg_kernel<<<BATCH, 128, 0, stream>>>(agg_w, ev, out);
}